// CorrelationSampler_62319975465628
// MI455X (gfx1250) — compile-verified
//
#include <hip/hip_runtime.h>

// Correlation sampler (RAFT-style bilinear lookup), MI455X / gfx1250.
// Bandwidth-bound gather: ~0.5 GB effective HBM traffic -> ~22 us floor @ 23.3 TB/s.
// WMMA is not applicable (per-pixel distinct B-operand); optimize data movement:
//   - coalesced b128 loads of the 4 gathered correlation rows (RT hint, L2 reuse)
//   - non-temporal b128 stores of the write-once 256 MB output
//   - global_prefetch_b8 for the next channel chunk (gfx1250 prefetch path)

typedef float v4f __attribute__((ext_vector_type(4)));

constexpr int Bn = 4;
constexpr int Hn = 64;
constexpr int Wn = 64;
constexpr int Cn = Hn * Wn;        // 4096 channels per pixel
constexpr int THREADS = 256;       // 8 wave32
constexpr int STRIDE = THREADS * 4; // floats advanced per loop iteration
constexpr int ITERS = Cn / STRIDE;  // 4

__global__ __launch_bounds__(THREADS)
void corr_sampler_kernel(const float* __restrict__ corr,
                         const float* __restrict__ flow,
                         float* __restrict__ out)
{
    const int pixel = blockIdx.x;            // b*H*W + h*W + w
    const int px = pixel & (Wn - 1);
    const int py = (pixel >> 6) & (Hn - 1);
    const int b  = pixel >> 12;

    // flow[b, 0, py, px], flow[b, 1, py, px] -- uniform across block (scalar loads)
    const int hw = py * Wn + px;
    const float fx = flow[((size_t)b * 2 + 0) * (Hn * Wn) + hw];
    const float fy = flow[((size_t)b * 2 + 1) * (Hn * Wn) + hw];

    // align_corners=True + border clamp reduces to: pos + flow*(N-1)/N, clipped.
    float ix = fminf(fmaxf((float)px + fx * ((float)(Wn - 1) / (float)Wn), 0.0f), (float)(Wn - 1));
    float iy = fminf(fmaxf((float)py + fy * ((float)(Hn - 1) / (float)Hn), 0.0f), (float)(Hn - 1));

    const float ix0f = floorf(ix);
    const float iy0f = floorf(iy);
    const float wx = ix - ix0f;
    const float wy = iy - iy0f;

    int ix0 = min(max((int)ix0f, 0), Wn - 1);
    int iy0 = min(max((int)iy0f, 0), Hn - 1);
    int ix1 = min(ix0 + 1, Wn - 1);
    int iy1 = min(iy0 + 1, Hn - 1);

    const float w00 = (1.0f - wy) * (1.0f - wx);
    const float w01 = (1.0f - wy) * wx;
    const float w10 = wy * (1.0f - wx);
    const float w11 = wy * wx;

    const float* __restrict__ base = corr + (size_t)b * Hn * Wn * Cn;
    const float* __restrict__ r00 = base + ((size_t)iy0 * Wn + ix0) * Cn;
    const float* __restrict__ r01 = base + ((size_t)iy0 * Wn + ix1) * Cn;
    const float* __restrict__ r10 = base + ((size_t)iy1 * Wn + ix0) * Cn;
    const float* __restrict__ r11 = base + ((size_t)iy1 * Wn + ix1) * Cn;

    float* __restrict__ o = out + (size_t)pixel * Cn;

    int c = (int)threadIdx.x * 4;

    #pragma unroll
    for (int i = 0; i < ITERS; ++i, c += STRIDE) {
        if (i + 1 < ITERS) {
            // gfx1250 prefetch path: warm next chunk of all four gathered rows.
            __builtin_prefetch(r00 + c + STRIDE, 0, 3);
            __builtin_prefetch(r01 + c + STRIDE, 0, 3);
            __builtin_prefetch(r10 + c + STRIDE, 0, 3);
            __builtin_prefetch(r11 + c + STRIDE, 0, 3);
        }
        v4f a = *(const v4f*)(r00 + c);
        v4f bv = *(const v4f*)(r01 + c);
        v4f cv = *(const v4f*)(r10 + c);
        v4f dv = *(const v4f*)(r11 + c);
        v4f acc = a * w00 + bv * w01 + cv * w10 + dv * w11;
        // Output is write-once/streamed: non-temporal store keeps L2 for correlation.
        __builtin_nontemporal_store(acc, (v4f*)(o + c));
    }
}

extern "C" void kernel_launch(void* const* d_in, const int* in_sizes, int n_in,
                              void* d_out, int out_size, void* d_ws, size_t ws_size,
                              hipStream_t stream) {
    const float* corr = (const float*)d_in[0]; // [B, H, W, H, W] fp32
    const float* flow = (const float*)d_in[1]; // [B, 2, H, W]    fp32
    float* out = (float*)d_out;                // [B, H, W, H*W]  fp32

    const int nblocks = Bn * Hn * Wn;          // 16384 pixels, one block each
    corr_sampler_kernel<<<nblocks, THREADS, 0, stream>>>(corr, flow, out);
}